// PLE_6227702579412
// MI455X (gfx1250) — compile-verified
//
#include <hip/hip_runtime.h>
#include <hip/hip_bf16.h>

typedef float v2f __attribute__((ext_vector_type(2)));
typedef float v8f __attribute__((ext_vector_type(8)));
typedef long long ll;

#define EPSBN 1e-5f

// -------------------------------------------------------------------------
// fp32 WMMA: D(16x16) = A(16x4) * B(4x16) + C, f32 everywhere.
// A layout (ISA 7.12.2): lane 0-15 row=lane, v0=K0,v1=K1; lane16-31 v0=K2,v1=K3.
// B layout (mirror):     lane 0-15 col=lane, v0=K0,v1=K1; lane16-31 v0=K2,v1=K3.
// C/D layout: vgpr v, lanes 0-15: (row=v, col=lane); lanes16-31: (row=v+8,col=lane-16)
// -------------------------------------------------------------------------
__device__ __forceinline__ v8f wmma4(v2f a, v2f b, v8f c) {
  return __builtin_amdgcn_wmma_f32_16x16x4_f32(false, a, false, b, (short)0, c,
                                               false, false);
}

// CDNA5 async DMA: memory -> LDS, 16B per lane, tracked by ASYNCcnt.
// saddr form: dsaddr from VGPR (32b LDS byte address), global = SGPR64 + VGPR32.
__device__ __forceinline__ void async_b128(unsigned lds, unsigned goff,
                                           unsigned long long base) {
  asm volatile("global_load_async_to_lds_b128 %0, %1, %2"
               :: "v"(lds), "v"(goff), "s"(base)
               : "memory");
}
__device__ __forceinline__ void wait_async0() {
  asm volatile("s_wait_asynccnt 0" ::: "memory");
}

// ---------------- batched GEMM: C[e] = A[sel(e)] @ W[e] + bias[e] --------
// amode: 0 = all experts share A[0]; 1 = identity A[e]; 2 = PLE map
// (e<16 -> e/2, else 8).  M,N multiples of 128; K multiple of 32.
#define BM 128
#define BNT 128
#define BKT 32

__global__ __launch_bounds__(256) void gemm_wmma_f32(
    const float* __restrict__ A, ll strideA, int amode,
    const float* __restrict__ W, ll strideW,
    const float* __restrict__ bias, ll strideBias,
    float* __restrict__ C, ll strideC,
    int M, int N, int K) {
  // ping-pong tiles; row strides 36/132 floats (16B aligned, bank-spread)
  __shared__ __align__(16) float As[2][BM][BKT + 4];
  __shared__ __align__(16) float Bs[2][BKT][BNT + 4];

  const int e = blockIdx.z;
  int asel;
  if (amode == 0)      asel = 0;
  else if (amode == 1) asel = e;
  else                 asel = (e < 16) ? (e >> 1) : 8;

  const float* Ae = A + (ll)asel * strideA;
  const float* We = W + (ll)e * strideW;
  const float* be = bias + (ll)e * strideBias;
  float* Ce = C + (ll)e * strideC;
  const unsigned long long AeU = (unsigned long long)(uintptr_t)Ae;
  const unsigned long long WeU = (unsigned long long)(uintptr_t)We;

  const int tid   = threadIdx.x;
  const int wave  = tid >> 5;
  const int lane  = tid & 31;
  const int lhalf = lane >> 4;
  const int lid   = lane & 15;
  const int waveM = wave >> 2;  // 0..1  (64 rows each)
  const int waveN = wave & 3;   // 0..3  (32 cols each)

  const int rowBase = blockIdx.y * BM;
  const int colBase = blockIdx.x * BNT;

  v8f acc[4][2];
  const v8f vz = {0.f, 0.f, 0.f, 0.f, 0.f, 0.f, 0.f, 0.f};
#pragma unroll
  for (int tm = 0; tm < 4; ++tm)
#pragma unroll
    for (int tn = 0; tn < 2; ++tn) acc[tm][tn] = vz;

  const int ar = tid >> 3;        // 0..31  (A tile: 32 rows/step, 8 thr/row)
  const int ac = (tid & 7) << 2;  // 0..28
  const int brr = tid >> 5;       // 0..7   (B tile: 8 rows/step, 32 thr/row)
  const int bc = (tid & 31) << 2; // 0..124

  // issue one 128x32 A tile + 32x128 B tile into buffer `buf` (async DMA)
  auto issue_tile = [&](int buf, int k0) {
#pragma unroll
    for (int s = 0; s < 4; ++s) {
      const int row = ar + s * 32;
      const unsigned ldsa = (unsigned)(uintptr_t)&As[buf][row][ac];
      const unsigned goff = (unsigned)(((ll)(rowBase + row) * K + (k0 + ac)) * 4);
      async_b128(ldsa, goff, AeU);
    }
#pragma unroll
    for (int s = 0; s < 4; ++s) {
      const int row = brr + s * 8;
      const unsigned ldsb = (unsigned)(uintptr_t)&Bs[buf][row][bc];
      const unsigned goff = (unsigned)(((ll)(k0 + row) * N + (colBase + bc)) * 4);
      async_b128(ldsb, goff, WeU);
    }
  };

  issue_tile(0, 0);
  wait_async0();
  __syncthreads();

  int cur = 0;
  for (int k0 = 0; k0 < K; k0 += BKT) {
    const int nxt = cur ^ 1;
    if (k0 + BKT < K) issue_tile(nxt, k0 + BKT);  // prefetch overlaps WMMAs

#pragma unroll
    for (int kk = 0; kk < BKT; kk += 4) {
      v2f afr[4], bfr[2];
#pragma unroll
      for (int tm = 0; tm < 4; ++tm) {
        const int r = waveM * 64 + tm * 16 + lid;
        afr[tm].x = As[cur][r][kk + 2 * lhalf];
        afr[tm].y = As[cur][r][kk + 2 * lhalf + 1];
      }
#pragma unroll
      for (int tn = 0; tn < 2; ++tn) {
        const int c = waveN * 32 + tn * 16 + lid;
        bfr[tn].x = Bs[cur][kk + 2 * lhalf][c];
        bfr[tn].y = Bs[cur][kk + 2 * lhalf + 1][c];
      }
#pragma unroll
      for (int tm = 0; tm < 4; ++tm)
#pragma unroll
        for (int tn = 0; tn < 2; ++tn)
          acc[tm][tn] = wmma4(afr[tm], bfr[tn], acc[tm][tn]);
    }

    wait_async0();     // next tile's DMA (this wave's share) is in LDS
    __syncthreads();   // all waves' shares visible; cur buffer free for reuse
    cur = nxt;
  }

#pragma unroll
  for (int tm = 0; tm < 4; ++tm)
#pragma unroll
    for (int tn = 0; tn < 2; ++tn) {
      const int col = colBase + waveN * 32 + tn * 16 + lid;
      const float bv = be[col];
#pragma unroll
      for (int v = 0; v < 8; ++v) {
        const int row = rowBase + waveM * 64 + tm * 16 + v + 8 * lhalf;
        Ce[(ll)row * N + col] = acc[tm][tn][v] + bv;
      }
    }
}

// --------------- BatchNorm (training-mode, biased var) + ReLU, in place --
// One thread per channel; stats over the batch dim then apply. Coalesced.
__global__ void bn_relu_kernel(float* __restrict__ H,
                               const float* __restrict__ gamma,
                               const float* __restrict__ beta,
                               int rows, int ncols) {
  const int e = blockIdx.y;
  const int j = blockIdx.x * blockDim.x + threadIdx.x;
  if (j >= ncols) return;
  float* col = H + (ll)e * rows * ncols + j;
  const float g  = gamma[(ll)e * ncols + j];
  const float bb = beta[(ll)e * ncols + j];
  float s = 0.f, s2 = 0.f;
  for (int b = 0; b < rows; ++b) {
    const float v = col[(ll)b * ncols];
    s += v; s2 += v * v;
  }
  const float m   = s / (float)rows;
  const float var = s2 / (float)rows - m * m;
  const float inv = rsqrtf(var + EPSBN) * g;
  for (int b = 0; b < rows; ++b) {
    const float v = col[(ll)b * ncols];
    const float r = (v - m) * inv + bb;
    col[(ll)b * ncols] = r > 0.f ? r : 0.f;
  }
}

// --------------- small gate linear: Out[g][b][k] = X[sel]·W[g][:,k]+b ----
__global__ void gate_linear_kernel(const float* __restrict__ X, ll strideX, int amode,
                                   const float* __restrict__ W, ll strideW,
                                   const float* __restrict__ bias, ll strideBias,
                                   float* __restrict__ Out, ll strideOut,
                                   int B, int in_dim, int N) {
  const int g = blockIdx.y;
  const ll idx = (ll)blockIdx.x * blockDim.x + threadIdx.x;
  if (idx >= (ll)B * N) return;
  const int b = (int)(idx / N);
  const int k = (int)(idx % N);
  const float* x = X + (amode == 0 ? 0 : (ll)g * strideX) + (ll)b * in_dim;
  const float* w = W + (ll)g * strideW + k;
  float acc = bias[(ll)g * strideBias + k];
  for (int i = 0; i < in_dim; ++i) acc += x[i] * w[(ll)i * N];
  Out[(ll)g * strideOut + (ll)b * N + k] = acc;
}

// --------------- per-channel stats for gate BN (channel = g*N + k) -------
__global__ void gate_stats_kernel(const float* __restrict__ Gh, int B, int N,
                                  float* __restrict__ mean, float* __restrict__ inv) {
  const int ch = blockIdx.x;
  const int g = ch / N, k = ch % N;
  const float* base = Gh + (ll)g * B * N + k;
  float s = 0.f, s2 = 0.f;
  for (int b = threadIdx.x; b < B; b += blockDim.x) {
    const float v = base[(ll)b * N];
    s += v; s2 += v * v;
  }
  __shared__ float sh[256], sh2[256];
  sh[threadIdx.x] = s; sh2[threadIdx.x] = s2;
  __syncthreads();
  for (int off = 128; off > 0; off >>= 1) {
    if ((int)threadIdx.x < off) {
      sh[threadIdx.x]  += sh[threadIdx.x + off];
      sh2[threadIdx.x] += sh2[threadIdx.x + off];
    }
    __syncthreads();
  }
  if (threadIdx.x == 0) {
    const float m = sh[0] / (float)B;
    const float var = sh2[0] / (float)B - m * m;
    mean[ch] = m;
    inv[ch]  = rsqrtf(var + EPSBN);
  }
}

// --------------- BN-apply + softmax over last axis (N <= 18) -------------
__global__ void gate_softmax_kernel(const float* __restrict__ Gh,
                                    const float* __restrict__ mean,
                                    const float* __restrict__ inv,
                                    const float* __restrict__ gamma,
                                    const float* __restrict__ beta,
                                    float* __restrict__ Out, int B, int N) {
  const int g = blockIdx.y;
  const ll b = (ll)blockIdx.x * blockDim.x + threadIdx.x;
  if (b >= B) return;
  const float* src = Gh + (ll)g * B * N + b * N;
  float vals[18];
  float mx = -3.4e38f;
  for (int k = 0; k < N; ++k) {
    const int c = g * N + k;
    const float v = (src[k] - mean[c]) * inv[c] * gamma[c] + beta[c];
    vals[k] = v;
    mx = v > mx ? v : mx;
  }
  float sum = 0.f;
  for (int k = 0; k < N; ++k) { vals[k] = __expf(vals[k] - mx); sum += vals[k]; }
  const float r = 1.f / sum;
  float* dst = Out + (ll)g * B * N + b * N;
  for (int k = 0; k < N; ++k) dst[k] = vals[k] * r;
}

// --------------- CGC mixture: Out[d][b][h] = sum_k gate*eo[sel(d,k)] -----
// K=4 hard-coded: sel = k<2 ? 2d+k : 16+(k-2)
__global__ void combine_kernel(const float* __restrict__ Gates,
                               const float* __restrict__ EO,
                               float* __restrict__ Out, int B, int Hd) {
  const int d = blockIdx.y;
  const ll idx = (ll)blockIdx.x * blockDim.x + threadIdx.x;
  const int h4cnt = Hd >> 2;
  if (idx >= (ll)B * h4cnt) return;
  const int b  = (int)(idx / h4cnt);
  const int h4 = (int)(idx % h4cnt) << 2;
  const float* gp = Gates + ((ll)d * B + b) * 4;
  float4 acc = {0.f, 0.f, 0.f, 0.f};
#pragma unroll
  for (int k = 0; k < 4; ++k) {
    const int e = (k < 2) ? (d * 2 + k) : (16 + (k - 2));
    const float gk = gp[k];
    const float4 ev = *(const float4*)(EO + ((ll)e * B + b) * Hd + h4);
    acc.x += gk * ev.x; acc.y += gk * ev.y; acc.z += gk * ev.z; acc.w += gk * ev.w;
  }
  *(float4*)(Out + ((ll)d * B + b) * Hd + h4) = acc;
}

// --------------- shared-gate pool: Out[b][h] = sum_e gsh[b][e]*eo[e][b][h]
__global__ void pooled_kernel(const float* __restrict__ Gsh,
                              const float* __restrict__ EO,
                              float* __restrict__ Out, int B, int Hd) {
  const ll idx = (ll)blockIdx.x * blockDim.x + threadIdx.x;
  const int h4cnt = Hd >> 2;
  if (idx >= (ll)B * h4cnt) return;
  const int b  = (int)(idx / h4cnt);
  const int h4 = (int)(idx % h4cnt) << 2;
  const float* gp = Gsh + (ll)b * 18;
  float4 acc = {0.f, 0.f, 0.f, 0.f};
#pragma unroll
  for (int e = 0; e < 18; ++e) {
    const float gk = gp[e];
    const float4 ev = *(const float4*)(EO + ((ll)e * B + b) * Hd + h4);
    acc.x += gk * ev.x; acc.y += gk * ev.y; acc.z += gk * ev.z; acc.w += gk * ev.w;
  }
  *(float4*)(Out + (ll)b * Hd + h4) = acc;
}

// --------------- tower head: dot(256) + bias, sigmoid, domain select -----
__global__ void tower_out_kernel(const float* __restrict__ Th,
                                 const float* __restrict__ W2,
                                 const float* __restrict__ b2,
                                 const int* __restrict__ domain,
                                 float* __restrict__ out, int B) {
  const ll idx = (ll)blockIdx.x * blockDim.x + threadIdx.x;
  if (idx >= (ll)B * 2) return;
  const int b = (int)(idx >> 1);
  const int t = (int)(idx & 1);
  const int d = t * 4 + domain[b];
  const float* h = Th + ((ll)d * B + b) * 256;
  const float* w = W2 + (ll)d * 256;
  float acc = b2[d];
  for (int c = 0; c < 256; ++c) acc += h[c] * w[c];
  out[(ll)b * 2 + t] = 1.f / (1.f + __expf(-acc));
}

// =========================================================================
extern "C" void kernel_launch(void* const* d_in, const int* in_sizes, int n_in,
                              void* d_out, int out_size, void* d_ws, size_t ws_size,
                              hipStream_t stream) {
  // ---- model dims ----
  const int Bn = 4096, D_IN = 1024, H1n = 1024, H2n = 512, En = 18, Kg = 4,
            DNn = 8, THn = 256;

  // ---- input order detection (insertion-order vs jax-sorted pytree) ----
  int iEmb = 0, iDom = 1;
  if (in_sizes[0] == 4096) { iDom = 0; iEmb = 1; }   // sorted: domain_id first
  const int pbase = 2;
  const bool srt = (in_sizes[pbase + 1] == 9437184); // sorted => W2 follows W1

  struct LevelIdx { int W1, b1, g1, bb1, W2, b2, g2, bb2, gW, gB, gG, gBB, sW, sB, sG, sBB; };
  LevelIdx L[2];
  for (int l = 0; l < 2; ++l) {
    const int o = pbase + l * 16;
    if (srt) {
      // experts: W1,W2,b1,b2,bb1,bb2,g1,g2 | gate_shared: W,b,bb,g | gates: W,b,bb,g
      L[l] = {o + 0, o + 2, o + 6, o + 4, o + 1, o + 3, o + 7, o + 5,
              o + 12, o + 13, o + 15, o + 14,
              o + 8, o + 9, o + 11, o + 10};
    } else {
      // experts: W1,b1,g1,bb1,W2,b2,g2,bb2 | gates: W,b,g,bb | gate_shared: W,b,g,bb
      L[l] = {o + 0, o + 1, o + 2, o + 3, o + 4, o + 5, o + 6, o + 7,
              o + 8, o + 9, o + 10, o + 11,
              o + 12, o + 13, o + 14, o + 15};
    }
  }
  struct TowerIdx { int W1, b1, g1, bb1, W2, b2; } T;
  {
    const int o = pbase + 32;
    if (srt) T = {o + 0, o + 2, o + 5, o + 4, o + 1, o + 3}; // W1,W2,b1,b2,bb1,g1
    else     T = {o + 0, o + 1, o + 2, o + 3, o + 4, o + 5};
  }
  auto P = [&](int i) { return (const float*)d_in[i]; };
  const float* embed  = (const float*)d_in[iEmb];
  const int*   domain = (const int*)d_in[iDom];

  // ---- workspace carve (floats) ----
  float* ws = (float*)d_ws;
  float* Hbig   = ws;                         // 18*4096*1024
  float* EO     = Hbig + 75497472ll;          // 18*4096*512
  float* X1     = EO + 37748736ll;            // 9*4096*512
  float* X2     = X1 + 18874368ll;            // 8*4096*512
  float* TowerH = X2 + 16777216ll;            // 8*4096*256
  float* Gh     = TowerH + 8388608ll;         // 8*4096*4
  float* Gates  = Gh + 131072ll;              // 8*4096*4
  float* GshH   = Gates + 131072ll;           // 4096*18
  float* Gsh    = GshH + 73728ll;             // 4096*18
  float* Gm     = Gsh + 73728ll;              // 64
  float* Gi     = Gm + 64;                    // 64
  float* GshM   = Gi + 64;                    // 32
  float* GshI   = GshM + 32;                  // 32

  const dim3 blk(256);

  // ================= Level 0 (in_dim = 1024) =================
  gemm_wmma_f32<<<dim3(H1n / 128, Bn / 128, En), blk, 0, stream>>>(
      embed, 0, /*shared*/0, P(L[0].W1), (ll)D_IN * H1n, P(L[0].b1), H1n,
      Hbig, (ll)Bn * H1n, Bn, H1n, D_IN);
  bn_relu_kernel<<<dim3(H1n / 256, En), blk, 0, stream>>>(
      Hbig, P(L[0].g1), P(L[0].bb1), Bn, H1n);
  gemm_wmma_f32<<<dim3(H2n / 128, Bn / 128, En), blk, 0, stream>>>(
      Hbig, (ll)Bn * H1n, /*ident*/1, P(L[0].W2), (ll)H1n * H2n, P(L[0].b2), H2n,
      EO, (ll)Bn * H2n, Bn, H2n, H1n);
  bn_relu_kernel<<<dim3(H2n / 256, En), blk, 0, stream>>>(
      EO, P(L[0].g2), P(L[0].bb2), Bn, H2n);

  gate_linear_kernel<<<dim3((Bn * Kg + 255) / 256, DNn), blk, 0, stream>>>(
      embed, 0, 0, P(L[0].gW), (ll)D_IN * Kg, P(L[0].gB), Kg,
      Gh, (ll)Bn * Kg, Bn, D_IN, Kg);
  gate_stats_kernel<<<dim3(DNn * Kg), blk, 0, stream>>>(Gh, Bn, Kg, Gm, Gi);
  gate_softmax_kernel<<<dim3(Bn / 256, DNn), blk, 0, stream>>>(
      Gh, Gm, Gi, P(L[0].gG), P(L[0].gBB), Gates, Bn, Kg);

  gate_linear_kernel<<<dim3((Bn * En + 255) / 256, 1), blk, 0, stream>>>(
      embed, 0, 0, P(L[0].sW), (ll)D_IN * En, P(L[0].sB), En,
      GshH, (ll)Bn * En, Bn, D_IN, En);
  gate_stats_kernel<<<dim3(En), blk, 0, stream>>>(GshH, Bn, En, GshM, GshI);
  gate_softmax_kernel<<<dim3(Bn / 256, 1), blk, 0, stream>>>(
      GshH, GshM, GshI, P(L[0].sG), P(L[0].sBB), Gsh, Bn, En);

  combine_kernel<<<dim3((Bn * (H2n / 4) + 255) / 256, DNn), blk, 0, stream>>>(
      Gates, EO, X1, Bn, H2n);
  pooled_kernel<<<dim3((Bn * (H2n / 4) + 255) / 256), blk, 0, stream>>>(
      Gsh, EO, X1 + 8ll * Bn * H2n, Bn, H2n);

  // ================= Level 1 (in_dim = 512) =================
  gemm_wmma_f32<<<dim3(H1n / 128, Bn / 128, En), blk, 0, stream>>>(
      X1, (ll)Bn * H2n, /*PLE map*/2, P(L[1].W1), (ll)H2n * H1n, P(L[1].b1), H1n,
      Hbig, (ll)Bn * H1n, Bn, H1n, H2n);
  bn_relu_kernel<<<dim3(H1n / 256, En), blk, 0, stream>>>(
      Hbig, P(L[1].g1), P(L[1].bb1), Bn, H1n);
  gemm_wmma_f32<<<dim3(H2n / 128, Bn / 128, En), blk, 0, stream>>>(
      Hbig, (ll)Bn * H1n, 1, P(L[1].W2), (ll)H1n * H2n, P(L[1].b2), H2n,
      EO, (ll)Bn * H2n, Bn, H2n, H1n);
  bn_relu_kernel<<<dim3(H2n / 256, En), blk, 0, stream>>>(
      EO, P(L[1].g2), P(L[1].bb2), Bn, H2n);

  gate_linear_kernel<<<dim3((Bn * Kg + 255) / 256, DNn), blk, 0, stream>>>(
      X1, (ll)Bn * H2n, /*ident*/1, P(L[1].gW), (ll)H2n * Kg, P(L[1].gB), Kg,
      Gh, (ll)Bn * Kg, Bn, H2n, Kg);
  gate_stats_kernel<<<dim3(DNn * Kg), blk, 0, stream>>>(Gh, Bn, Kg, Gm, Gi);
  gate_softmax_kernel<<<dim3(Bn / 256, DNn), blk, 0, stream>>>(
      Gh, Gm, Gi, P(L[1].gG), P(L[1].gBB), Gates, Bn, Kg);

  combine_kernel<<<dim3((Bn * (H2n / 4) + 255) / 256, DNn), blk, 0, stream>>>(
      Gates, EO, X2, Bn, H2n);

  // ================= Towers =================
  gemm_wmma_f32<<<dim3(THn / 128, Bn / 128, DNn), blk, 0, stream>>>(
      X2, (ll)Bn * H2n, 1, P(T.W1), (ll)H2n * THn, P(T.b1), THn,
      TowerH, (ll)Bn * THn, Bn, THn, H2n);
  bn_relu_kernel<<<dim3(THn / 256, DNn), blk, 0, stream>>>(
      TowerH, P(T.g1), P(T.bb1), Bn, THn);
  tower_out_kernel<<<dim3((Bn * 2 + 255) / 256), blk, 0, stream>>>(
      TowerH, P(T.W2), P(T.b2), domain, (float*)d_out, Bn);
}